// TimeSeriesOps_15401752724188
// MI455X (gfx1250) — compile-verified
//
#include <hip/hip_runtime.h>
#include <math.h>

// Rolling mean (w=256) over T axis of (T=4096, N=512, J=16) f32, reproducing
// the reference's block-overlap scheme (block_size=512, per-block sum reset).
// Bandwidth-bound (~268 MB min traffic -> ~11.5us floor @ 23.3 TB/s).
// Per 16-row tile: D = L(16x16 lower-tri ones) * Y + broadcast(carry) via
// 4 chained V_WMMA_F32_16X16X4_F32; carry via ds_bpermute_b32.
// Y[r] = x[r] - x[r-w]; lag stream is L2-resident (192 MB L2 > 134 MB tensor).
// All boundary guards are tile-uniform scalar branches; steady state issues
// unguarded b32 loads/stores from one base VGPR pair + immediate offsets.

typedef float v2f __attribute__((ext_vector_type(2)));
typedef float v8f __attribute__((ext_vector_type(8)));

#define T_DIM     4096
#define COLS      8192      // N*J, contiguous fastest axis
#define W         256
#define BLOCK_OUT 512
#define NTILES    48        // covers w-1+BLOCK_OUT = 767 rows

__device__ __forceinline__ float clean(float v) {
    // nan_to_num: NaN -> 0, +/-inf -> +/-FLT_MAX
    v = (v != v) ? 0.0f : v;
    return fminf(fmaxf(v, -3.4028235e38f), 3.4028235e38f);
}

__device__ __forceinline__ float ld_guard(const float* __restrict__ x, int g, int col) {
    return (g < T_DIM) ? clean(x[(size_t)g * COLS + col]) : 0.0f;  // zero pad past T
}

__global__ void __launch_bounds__(256)
roll_mean_wmma_f32(const float* __restrict__ x, float* __restrict__ out) {
    const int lane = threadIdx.x & 31;
    const int wave = threadIdx.x >> 5;
    const int half = lane >> 4;              // 0: lanes 0-15, 1: lanes 16-31
    const int m    = lane & 15;              // A row (per half) == B/C/D column N
    const int col  = (blockIdx.x * 8 + wave) * 16 + m;
    const int bs   = blockIdx.y * BLOCK_OUT;

    // A = 16x16 lower-triangular ones as four 16x4 K-slices (lane constants).
    v2f a[4];
#pragma unroll
    for (int j = 0; j < 4; ++j) {
        const int k0 = 4 * j + 2 * half;
        a[j].x = (k0     <= m) ? 1.0f : 0.0f;
        a[j].y = (k0 + 1 <= m) ? 1.0f : 0.0f;
    }

    const float inv_w = 1.0f / (float)W;
    float carry = 0.0f;   // running column sum S (identical in both lane halves)

    for (int tile = 0; tile < NTILES; ++tile) {
        const int rbase = tile * 16;
        const int g0    = bs + rbase;        // first data row of this tile

        // Prefetch leading stream 4 tiles ahead (global_prefetch_b8), clamped.
        {
            int gp = g0 + 64;
            if (gp >= T_DIM) gp = T_DIM - 1;
            __builtin_prefetch(x + (size_t)gp * COLS + col, 0, 0);
        }

        // B slices: rows rbase + 4j + 2*half + {0,1} of Y = x - lag(x, w).
        v2f b[4];
        if (g0 + 16 <= T_DIM) {
            // Uniform fast path: whole tile in range; one base, imm offsets.
            const float* q  = x + (size_t)(g0 + 2 * half) * COLS + col;
            const float* ql = q - (size_t)W * COLS;       // lag rows
#pragma unroll
            for (int j = 0; j < 4; ++j) {
                float v0 = clean(q[(4 * j)     * COLS]);
                float v1 = clean(q[(4 * j + 1) * COLS]);
                if (tile >= 16) {             // uniform: rows >= W here
                    v0 -= clean(ql[(4 * j)     * COLS]);
                    v1 -= clean(ql[(4 * j + 1) * COLS]);
                }
                b[j].x = v0; b[j].y = v1;
            }
        } else {
            // Zero-padded tail of last time block (tiles >= 32 => rows >= W).
#pragma unroll
            for (int j = 0; j < 4; ++j) {
                const int r0 = rbase + 4 * j + 2 * half;
                float v0 = ld_guard(x, bs + r0,     col) - ld_guard(x, bs + r0 - W,     col);
                float v1 = ld_guard(x, bs + r0 + 1, col) - ld_guard(x, bs + r0 + 1 - W, col);
                b[j].x = v0; b[j].y = v1;
            }
        }

        // D = L*Y + broadcast(carry): 4 chained f32 WMMAs (K=4 each).
        v8f acc = { carry, carry, carry, carry, carry, carry, carry, carry };
#pragma unroll
        for (int j = 0; j < 4; ++j)
            acc = __builtin_amdgcn_wmma_f32_16x16x4_f32(
                      false, a[j], false, b[j], (short)0, acc, false, false);

        // carry' = S[last tile row]: M=15 lives in VGPR7 of lanes 16-31.
        carry = __int_as_float(
            __builtin_amdgcn_ds_bpermute((m + 16) << 2, __float_as_int(acc[7])));

        if (tile == 15) {
            // First full-window value S[255]/w -> row bs+255; block 0 also
            // replicates it into rows 0..w-2 (reference's jnp.repeat).
            const float fv = carry * inv_w;
            if (half == 0)
                out[(size_t)(bs + 255) * COLS + col] = fv;
            if (blockIdx.y == 0) {
                float* p = out + (size_t)half * COLS + col;
                for (int t0 = 0; t0 < 254; t0 += 2) {     // rows 0..253, no guards
                    *p = fv;
                    p += 2 * COLS;
                }
                if (half == 0)                            // row 254 tail
                    out[(size_t)254 * COLS + col] = fv;
            }
        } else if (tile >= 16 && g0 < T_DIM) {
            float* p = out + (size_t)(g0 + 8 * half) * COLS + col;
            if (tile < 47 && g0 + 16 <= T_DIM) {
                // Uniform fast path: all 16 rows are valid outputs.
#pragma unroll
                for (int vi = 0; vi < 8; ++vi)
                    p[vi * COLS] = acc[vi] * inv_w;
            } else {
                // Partial tile: drop r=767 (tile 47) and rows past T.
#pragma unroll
                for (int vi = 0; vi < 8; ++vi) {
                    const int r = rbase + vi + 8 * half;
                    if (r <= 766 && bs + r < T_DIM)
                        p[vi * COLS] = acc[vi] * inv_w;
                }
            }
        }
    }
}

extern "C" void kernel_launch(void* const* d_in, const int* in_sizes, int n_in,
                              void* d_out, int out_size, void* d_ws, size_t ws_size,
                              hipStream_t stream) {
    (void)in_sizes; (void)n_in; (void)d_ws; (void)ws_size; (void)out_size;
    const float* x = (const float*)d_in[0];   // (4096, 512, 16) f32
    float* out     = (float*)d_out;           // (4096, 512, 16) f32
    // d_in[1]=window_size(256), d_in[2]=overlap_factor(2): fixed by setup_inputs.
    dim3 grid(COLS / 128, 8);                 // 64 col-groups x 8 time blocks
    roll_mean_wmma_f32<<<grid, 256, 0, stream>>>(x, out);
}